// LabelSmoothing_75866302317193
// MI455X (gfx1250) — compile-verified
//
#include <hip/hip_runtime.h>
#include <hip/hip_bf16.h>
#include <math.h>

#define SMOOTHING 0.1f
#define PAD_IDX   0
#define THREADS   256
#define WAVES     (THREADS / 32)
#define DEPTH     4                       // async pipeline depth per wave

// ---- CDNA5 async global->LDS helpers (gfx1250 ASYNCcnt path) ---------------
// Per-lane: copies 16 bytes from the global address in gptr to the LDS address in lds_off, tracked by ASYNCcnt.
// s_wait_dscnt 0 prefix guarantees the previous ds_load of the target buffer
// has fully completed before the async engine may overwrite it (WAR hazard on
// buffer reuse), without needing any workgroup barrier. th:TH_LOAD_NT because
// the 524 MB stream is read exactly once (> 192 MB L2) — don't thrash L2.
__device__ __forceinline__ void async_tile_load(unsigned lds_off, const float* gptr) {
    asm volatile(
        "s_wait_dscnt 0x0\n\t"
        "global_load_async_to_lds_b128 %0, %1, off th:TH_LOAD_NT"
        :: "v"(lds_off), "v"((unsigned long long)(size_t)gptr)
        : "memory");
}
__device__ __forceinline__ void wait_async_le(int n) {
    switch (n) {
    case 0:  asm volatile("s_wait_asynccnt 0x0" ::: "memory"); break;
    case 1:  asm volatile("s_wait_asynccnt 0x1" ::: "memory"); break;
    case 2:  asm volatile("s_wait_asynccnt 0x2" ::: "memory"); break;
    default: asm volatile("s_wait_asynccnt 0x3" ::: "memory"); break;
    }
}

// ---- Kernel 1: one workgroup per token -------------------------------------
__global__ __launch_bounds__(THREADS) void ls_token_kernel(
    const float* __restrict__ pred, const int* __restrict__ target,
    float* __restrict__ kld_out, int V) {
    __shared__ float4 abuf[WAVES * DEPTH * 32];   // per-wave ring buffers, 16 KB
    __shared__ float  wm[WAVES], wd[WAVES], wsp[WAVES];

    const int tid   = threadIdx.x;
    const int wave  = tid >> 5;
    const int lane  = tid & 31;
    const int token = blockIdx.x;
    const float* row = pred + (size_t)token * (size_t)V;

    const int tgt = target[token];
    float pt = 0.0f;
    if (tid == 0) pt = row[tgt];              // issue early; consumed at the end

    // online softmax state + plain sum
    float m = -INFINITY, d = 0.0f, sp = 0.0f;

    const int NT4 = V >> 10;                  // tiles of 1024 floats (256 lanes x float4)
    unsigned ldsb[DEPTH];
    #pragma unroll
    for (int b = 0; b < DEPTH; ++b)
        ldsb[b] = (unsigned)(size_t)&abuf[(wave * DEPTH + b) * 32 + lane];

    if (NT4 > 0) {
        const int pro = (NT4 < DEPTH) ? NT4 : DEPTH;
        for (int k = 0; k < pro; ++k)
            async_tile_load(ldsb[k], row + (size_t)k * 1024 + (size_t)tid * 4);
        for (int i = 0; i < NT4; ++i) {
            int rem = NT4 - 1 - i;            // tiles still outstanding beyond tile i
            wait_async_le(rem < (DEPTH - 1) ? rem : (DEPTH - 1));
            float4 x = abuf[(wave * DEPTH + (i % DEPTH)) * 32 + lane];
            float lm = fmaxf(fmaxf(x.x, x.y), fmaxf(x.z, x.w));
            float mn = fmaxf(m, lm);
            d = d * __expf(m - mn) + __expf(x.x - mn) + __expf(x.y - mn)
                                   + __expf(x.z - mn) + __expf(x.w - mn);
            m = mn;
            sp += (x.x + x.y) + (x.z + x.w);
            int next = i + DEPTH;             // refill the buffer just consumed
            if (next < NT4)
                async_tile_load(ldsb[i % DEPTH],
                                row + (size_t)next * 1024 + (size_t)tid * 4);
        }
    }
    // scalar tail (V=32000 -> exactly one element per thread, no divergence)
    for (int j = (NT4 << 10) + tid; j < V; j += THREADS) {
        float x  = row[j];
        float mn = fmaxf(m, x);
        d = d * __expf(m - mn) + __expf(x - mn);
        m = mn;
        sp += x;
    }

    // wave32 butterfly reduction (warpSize == 32 on gfx1250)
    for (int off = 16; off >= 1; off >>= 1) {
        float m2 = __shfl_xor(m, off);
        float d2 = __shfl_xor(d, off);
        float mn = fmaxf(m, m2);
        d = d * __expf(m - mn) + d2 * __expf(m2 - mn);
        m = mn;
        sp += __shfl_xor(sp, off);
    }
    if (lane == 0) { wm[wave] = m; wd[wave] = d; wsp[wave] = sp; }
    __syncthreads();

    if (tid == 0) {
        float M = wm[0], D = wd[0], SP = wsp[0];
        for (int w = 1; w < WAVES; ++w) {
            float mn = fmaxf(M, wm[w]);
            D = D * __expf(M - mn) + wd[w] * __expf(wm[w] - mn);
            M = mn;
            SP += wsp[w];
        }
        const double epsd = (double)SMOOTHING / (double)V;
        const double Td   = 1.0 - (double)SMOOTHING + epsd;
        const float  C    = (float)(epsd * (double)(V - 1) * log(epsd) + Td * log(Td));
        const float  eps  = (float)epsd;
        const float  LSE  = M + logf(D);
        const float  S_lp = SP - (float)V * LSE;                 // sum of log-probs
        const float  kld  = C - eps * S_lp - (1.0f - SMOOTHING) * (pt - LSE);
        kld_out[token] = (tgt != PAD_IDX) ? kld : 0.0f;
    }
}

// ---- Kernel 2: reduce per-token losses, divide by non-pad count ------------
__global__ __launch_bounds__(THREADS) void ls_finalize_kernel(
    const float* __restrict__ kld, const int* __restrict__ target,
    int ntok, float* __restrict__ out) {
    __shared__ float sk[WAVES], sc[WAVES];
    float k = 0.0f, c = 0.0f;
    for (int i = threadIdx.x; i < ntok; i += THREADS) {
        k += kld[i];
        c += (target[i] != PAD_IDX) ? 1.0f : 0.0f;
    }
    for (int off = 16; off >= 1; off >>= 1) {
        k += __shfl_xor(k, off);
        c += __shfl_xor(c, off);
    }
    const int wave = threadIdx.x >> 5, lane = threadIdx.x & 31;
    if (lane == 0) { sk[wave] = k; sc[wave] = c; }
    __syncthreads();
    if (threadIdx.x == 0) {
        float K = 0.0f, C2 = 0.0f;
        for (int w = 0; w < WAVES; ++w) { K += sk[w]; C2 += sc[w]; }
        out[0] = K / C2;
    }
}

extern "C" void kernel_launch(void* const* d_in, const int* in_sizes, int n_in,
                              void* d_out, int out_size, void* d_ws, size_t ws_size,
                              hipStream_t stream) {
    const float* pred   = (const float*)d_in[0];
    const int*   target = (const int*)d_in[1];
    float*       out    = (float*)d_out;
    const int ntok = in_sizes[1];                 // B*S
    const int V    = in_sizes[0] / ntok;          // vocab
    float* kld_ws = (float*)d_ws;                 // ntok floats of scratch

    ls_token_kernel<<<ntok, THREADS, 0, stream>>>(pred, target, kld_ws, V);
    ls_finalize_kernel<<<1, THREADS, 0, stream>>>(kld_ws, target, ntok, out);
}